// DPSR_30210799960070
// MI455X (gfx1250) — compile-verified
//
#include <hip/hip_runtime.h>

typedef float v2f __attribute__((ext_vector_type(2)));
typedef float v8f __attribute__((ext_vector_type(8)));

#define RESN 256
#define NVOX (RESN*RESN*RESN)
#define PI_F 3.14159265358979323846f

// ---------------------------------------------------------------------------
// DFT matrices, stored as split re/im planes in row-PAIR-interleaved layout:
//   plane[((k>>1)*256 + n)*2 + (k&1)] = M[k][n]
// so a WMMA B fragment (rows k,k+1 at column n) is a single b64 load that
// lands directly in the v2f operand — no register shuffles in the hot loop.
//   Wf = exp(-2*pi*i*k*n/256)            (forward)
//   Wi = exp(+2*pi*i*k*n/256) / 256      (inverse, 1/N folded per axis)
// angle reduced mod 256 exactly (k*n integer) -> full f32 accuracy.
// ---------------------------------------------------------------------------
__global__ void init_dft(float* __restrict__ WfRe, float* __restrict__ WfIm,
                         float* __restrict__ WiRe, float* __restrict__ WiIm) {
    int t = blockIdx.x * blockDim.x + threadIdx.x;   // 65536 threads
    int k = t >> 8, n = t & 255;
    int r = (k * n) & 255;
    float ang = -2.0f * PI_F * (float)r * (1.0f / 256.0f);
    float c = cosf(ang), s = sinf(ang);
    int idx = (((k >> 1) * 256) + n) * 2 + (k & 1);
    WfRe[idx] = c;
    WfIm[idx] = s;
    WiRe[idx] = c * (1.0f / 256.0f);
    WiIm[idx] = -s * (1.0f / 256.0f);
}

__global__ void zero_f32(float* __restrict__ p, int n) {
    int i = blockIdx.x * blockDim.x + threadIdx.x;
    if (i < n) p[i] = 0.0f;
}

// ---------------------------------------------------------------------------
// Trilinear scatter (point_rasterize): atomicAdd into 3-channel grid
// ---------------------------------------------------------------------------
__global__ void scatter_pts(const float* __restrict__ V, const float* __restrict__ N,
                            float* __restrict__ ras, int P) {
    int p = blockIdx.x * blockDim.x + threadIdx.x;
    if (p >= P) return;
    float xs = V[3*p+0] * 256.0f, ys = V[3*p+1] * 256.0f, zs = V[3*p+2] * 256.0f;
    float nx = N[3*p+0], ny = N[3*p+1], nz = N[3*p+2];
    int ix = (int)floorf(xs), iy = (int)floorf(ys), iz = (int)floorf(zs);
    float fx = xs - (float)ix, fy = ys - (float)iy, fz = zs - (float)iz;
    ix &= 255; iy &= 255; iz &= 255;
    #pragma unroll
    for (int c8 = 0; c8 < 8; ++c8) {
        int dx = (c8 >> 2) & 1, dy = (c8 >> 1) & 1, dz = c8 & 1;
        int jx = (ix + dx) & 255, jy = (iy + dy) & 255, jz = (iz + dz) & 255;
        float w = (dx ? fx : 1.0f - fx) * (dy ? fy : 1.0f - fy) * (dz ? fz : 1.0f - fz);
        int flat = (jx << 16) | (jy << 8) | jz;
        atomicAdd(&ras[flat],            nx * w);
        atomicAdd(&ras[NVOX + flat],     ny * w);
        atomicAdd(&ras[2*NVOX + flat],   nz * w);
    }
}

// ---------------------------------------------------------------------------
// One 1-D DFT pass along one axis, as a WMMA GEMM against the 256x256 DFT
// matrix. Each 512-thread block (16 waves) transforms a slab of 16 lines:
//   addr(m,k) = base + m*sm + k*sk,  base = outer*outerStride + slab*16*sm
// Slab staged through LDS (pitch 257 float2 -> conflict-free fragment reads),
// each wave owns one 16-wide N tile; K loop is 64 steps of 16x16x4 f32 WMMA.
// Complex GEMM: Cre += Are*Bre - Aim*Bim ; Cim += Are*Bim + Aim*Bre.
// In-place safe: a block only touches its own slab, all global reads precede
// all global writes (separated by the K loop + barriers).
// ---------------------------------------------------------------------------
template<bool REAL_IN, bool REAL_OUT>
__global__ void __launch_bounds__(512)
dft_pass(const float* __restrict__ rin, const float2* __restrict__ cin,
         float2* __restrict__ cout, float* __restrict__ rout,
         const float* __restrict__ Wre, const float* __restrict__ Wim,
         int sm, int sk, int outerStride) {
    __shared__ float sA[16 * 257 * 2];
    const int tid   = threadIdx.x;
    const int outer = blockIdx.x >> 4;
    const int slab  = blockIdx.x & 15;
    const int base  = outer * outerStride + slab * 16 * sm;
    const bool kfast = (sk == 1);

    // ---- stage slab into LDS, coalesced along the unit-stride dimension ----
    for (int it = 0; it < 8; ++it) {
        int idx = it * 512 + tid;                   // 16*256 complex elements
        int m = kfast ? (idx >> 8) : (idx & 15);
        int k = kfast ? (idx & 255) : (idx >> 4);
        int g = base + m * sm + k * sk;
        int l = (m * 257 + k) * 2;
        if (REAL_IN) { sA[l] = rin[g]; sA[l + 1] = 0.0f; }
        else         { float2 v = cin[g]; sA[l] = v.x; sA[l + 1] = v.y; }
    }
    __syncthreads();

    const int lane = tid & 31;
    const int wv   = tid >> 5;       // 16 waves -> 16 N tiles
    const int m    = lane & 15;      // A fragment row
    const int kh   = lane >> 4;      // K half select
    const int n    = lane & 15;      // B fragment column
    const int nt   = wv * 16;

    const v2f* __restrict__ WreP = (const v2f*)Wre;  // row-pair-interleaved
    const v2f* __restrict__ WimP = (const v2f*)Wim;

    v8f acc_re = {};
    v8f acc_im = {};

    for (int k0 = 0; k0 < 256; k0 += 4) {
        // A fragment 16x4: lane m, components K = k0+2*kh, k0+2*kh+1
        int la = (m * 257 + k0 + 2 * kh) * 2;
        v2f a_re = { sA[la],     sA[la + 2] };
        // B fragment 4x16: lane n, rows K = k0+2*kh, k0+2*kh+1.
        // k0+2*kh is even -> rows (k,k+1) form exactly one stored pair.
        int wb = ((k0 + 2 * kh) >> 1) * 256 + nt + n;
        v2f b_re = WreP[wb];
        v2f b_im = WimP[wb];

        acc_re = __builtin_amdgcn_wmma_f32_16x16x4_f32(false, a_re, false, b_re,
                                                       (short)0, acc_re, false, false);
        acc_im = __builtin_amdgcn_wmma_f32_16x16x4_f32(false, a_re, false, b_im,
                                                       (short)0, acc_im, false, false);
        if (!REAL_IN) {
            v2f a_im = { sA[la + 1], sA[la + 3] };
            v2f nai  = -a_im;      // f32 WMMA NEG bits are C-only -> negate in VALU
            acc_re = __builtin_amdgcn_wmma_f32_16x16x4_f32(false, nai,  false, b_im,
                                                           (short)0, acc_re, false, false);
            acc_im = __builtin_amdgcn_wmma_f32_16x16x4_f32(false, a_im, false, b_re,
                                                           (short)0, acc_im, false, false);
        }
    }
    __syncthreads();

    // ---- restage C tiles into LDS (lane writes row r+8*kh, col nt+n) ----
    #pragma unroll
    for (int r = 0; r < 8; ++r) {
        int mr  = r + 8 * kh;
        int l   = (mr * 257 + nt + n) * 2;
        sA[l]     = acc_re[r];
        sA[l + 1] = acc_im[r];
    }
    __syncthreads();

    // ---- coalesced writeback ----
    for (int it = 0; it < 8; ++it) {
        int idx = it * 512 + tid;
        int mm = kfast ? (idx >> 8) : (idx & 15);
        int kk = kfast ? (idx & 255) : (idx >> 4);
        int g = base + mm * sm + kk * sk;
        int l = (mm * 257 + kk) * 2;
        if (REAL_OUT) rout[g] = sA[l];
        else          cout[g] = make_float2(sA[l], sA[l + 1]);
    }
}

// ---------------------------------------------------------------------------
// acc += -i * omega_c * spec   (DivN accumulation, per channel)
// ---------------------------------------------------------------------------
__global__ void accum_divn(const float2* __restrict__ spec, float2* __restrict__ acc,
                           int channel, int init) {
    int i = blockIdx.x * blockDim.x + threadIdx.x;
    if (i >= NVOX) return;
    int kz = i & 255, ky = (i >> 8) & 255, kx = i >> 16;
    int k = (channel == 0) ? kx : ((channel == 1) ? ky : kz);
    float f = (k < 128) ? (float)k : (float)(k - 256);
    float om = 2.0f * PI_F * f;
    float2 v = spec[i];
    float ar =  om * v.y;          // (-i*om)*(re + i im) = om*im - i*om*re
    float ai = -om * v.x;
    if (init) acc[i] = make_float2(ar, ai);
    else { float2 a = acc[i]; acc[i] = make_float2(a.x + ar, a.y + ai); }
}

// Phi = DivN / (Lap + 1e-6),  Phi[0] = 0
__global__ void div_lap(float2* __restrict__ acc) {
    int i = blockIdx.x * blockDim.x + threadIdx.x;
    if (i >= NVOX) return;
    int kz = i & 255, ky = (i >> 8) & 255, kx = i >> 16;
    float fx = (kx < 128) ? (float)kx : (float)(kx - 256);
    float fy = (ky < 128) ? (float)ky : (float)(ky - 256);
    float fz = (kz < 128) ? (float)kz : (float)(kz - 256);
    float w2 = 4.0f * PI_F * PI_F * (fx * fx + fy * fy + fz * fz);
    float inv = 1.0f / (-w2 + 1e-6f);
    float2 a = acc[i];
    float2 r = make_float2(a.x * inv, a.y * inv);
    if (i == 0) r = make_float2(0.0f, 0.0f);
    acc[i] = r;
}

// scal[0] = 0 (gather accumulator);  scal[1] = phi[0,0,0]
__global__ void snag(float* __restrict__ scal, const float* __restrict__ phi) {
    scal[0] = 0.0f;
    scal[1] = phi[0];
}

// Trilinear gather of phi at the points; block-reduced sum -> scal[0]
__global__ void gather_mean(const float* __restrict__ V, const float* __restrict__ phi,
                            float* __restrict__ scal, int P) {
    __shared__ float red[256];
    int p = blockIdx.x * 256 + threadIdx.x;
    float fv = 0.0f;
    if (p < P) {
        float xs = V[3*p+0] * 256.0f, ys = V[3*p+1] * 256.0f, zs = V[3*p+2] * 256.0f;
        int ix = (int)floorf(xs), iy = (int)floorf(ys), iz = (int)floorf(zs);
        float fx = xs - (float)ix, fy = ys - (float)iy, fz = zs - (float)iz;
        ix &= 255; iy &= 255; iz &= 255;
        #pragma unroll
        for (int c8 = 0; c8 < 8; ++c8) {
            int dx = (c8 >> 2) & 1, dy = (c8 >> 1) & 1, dz = c8 & 1;
            int jx = (ix + dx) & 255, jy = (iy + dy) & 255, jz = (iz + dz) & 255;
            float w = (dx ? fx : 1.0f - fx) * (dy ? fy : 1.0f - fy) * (dz ? fz : 1.0f - fz);
            fv += w * phi[(jx << 16) | (jy << 8) | jz];
        }
    }
    red[threadIdx.x] = fv;
    __syncthreads();
    for (int s = 128; s > 0; s >>= 1) {
        if (threadIdx.x < s) red[threadIdx.x] += red[threadIdx.x + s];
        __syncthreads();
    }
    if (threadIdx.x == 0) atomicAdd(&scal[0], red[0]);
}

// out = -(out - offset) / |phi000 - offset| * 0.5
__global__ void finalize_phi(float* __restrict__ out, const float* __restrict__ scal,
                             float invP) {
    int i = blockIdx.x * blockDim.x + threadIdx.x;
    if (i >= NVOX) return;
    float offset = scal[0] * invP;
    float denom  = fabsf(scal[1] - offset);
    out[i] = -(out[i] - offset) / denom * 0.5f;
}

// ---------------------------------------------------------------------------
extern "C" void kernel_launch(void* const* d_in, const int* in_sizes, int n_in,
                              void* d_out, int out_size, void* d_ws, size_t ws_size,
                              hipStream_t stream) {
    const float* V = (const float*)d_in[0];
    const float* N = (const float*)d_in[1];
    float* out = (float*)d_out;
    char* ws = (char*)d_ws;

    float* WfRe = (float*)(ws);                        // 256 KB each
    float* WfIm = (float*)(ws + (size_t)(256 << 10));
    float* WiRe = (float*)(ws + (size_t)(512 << 10));
    float* WiIm = (float*)(ws + (size_t)(768 << 10));
    float* scal = (float*)(ws + (size_t)(1 << 20));    // 2 floats
    float* ras  = (float*)(ws + (size_t)(2 << 20));    // 3 * 64 MiB
    size_t rasBytes = (size_t)3 * NVOX * sizeof(float);
    float2* spec = (float2*)(ws + (size_t)(2 << 20) + rasBytes);           // 128 MiB
    float2* acc  = (float2*)((char*)spec + (size_t)NVOX * sizeof(float2)); // 128 MiB

    int P = in_sizes[0] / 3;

    init_dft<<<256, 256, 0, stream>>>(WfRe, WfIm, WiRe, WiIm);
    zero_f32<<<(3 * NVOX + 255) / 256, 256, 0, stream>>>(ras, 3 * NVOX);
    scatter_pts<<<(P + 255) / 256, 256, 0, stream>>>(V, N, ras, P);

    for (int c = 0; c < 3; ++c) {
        const float* rc = ras + (size_t)c * NVOX;
        // forward FFT: z (real->complex), then y, then x (both in place)
        dft_pass<true,  false><<<4096, 512, 0, stream>>>(rc, nullptr, spec, nullptr,
                                                         WfRe, WfIm, 256, 1, 65536);
        dft_pass<false, false><<<4096, 512, 0, stream>>>(nullptr, spec, spec, nullptr,
                                                         WfRe, WfIm, 1, 256, 65536);
        dft_pass<false, false><<<4096, 512, 0, stream>>>(nullptr, spec, spec, nullptr,
                                                         WfRe, WfIm, 1, 65536, 256);
        accum_divn<<<(NVOX + 255) / 256, 256, 0, stream>>>(spec, acc, c, c == 0);
    }

    div_lap<<<(NVOX + 255) / 256, 256, 0, stream>>>(acc);

    // inverse FFT (Wi carries conj + 1/256 per pass): z, y in place, x -> real out
    dft_pass<false, false><<<4096, 512, 0, stream>>>(nullptr, acc, acc, nullptr,
                                                     WiRe, WiIm, 256, 1, 65536);
    dft_pass<false, false><<<4096, 512, 0, stream>>>(nullptr, acc, acc, nullptr,
                                                     WiRe, WiIm, 1, 256, 65536);
    dft_pass<false, true ><<<4096, 512, 0, stream>>>(nullptr, acc, nullptr, out,
                                                     WiRe, WiIm, 1, 65536, 256);

    snag<<<1, 1, 0, stream>>>(scal, out);
    gather_mean<<<(P + 255) / 256, 256, 0, stream>>>(V, out, scal, P);
    finalize_phi<<<(NVOX + 255) / 256, 256, 0, stream>>>(out, scal, 1.0f / (float)P);
}